// Model_1778116460894
// MI455X (gfx1250) — compile-verified
//
#include <hip/hip_runtime.h>

typedef __attribute__((ext_vector_type(2))) float v2f;
typedef __attribute__((ext_vector_type(8))) float v8f;

#define NNODES 50000
#define NEDGES 800000
#define NFIN   64
#define FILT   32
#define NHEADS 4
#define HDIM   128   // NHEADS * FILT
#define OUTLEN 12

// ---------------------------------------------------------------------------
// init: deg = 1 (self loop), smax = -inf, denom = 0
// ---------------------------------------------------------------------------
__global__ void k_init(float* __restrict__ deg, float* __restrict__ smax,
                       float* __restrict__ denom) {
    int i = blockIdx.x * blockDim.x + threadIdx.x;
    if (i < NNODES) deg[i] = 1.0f;
    if (i < NNODES * NHEADS) {
        smax[i]  = -__builtin_inff();
        denom[i] = 0.0f;
    }
}

// ---------------------------------------------------------------------------
// degree accumulation over destination (col), then dinv = rsqrt(deg)
// ---------------------------------------------------------------------------
__global__ void k_degree(const int* __restrict__ col, float* __restrict__ deg) {
    int e = blockIdx.x * blockDim.x + threadIdx.x;
    if (e < NEDGES) atomicAdd(&deg[col[e]], 1.0f);
}

__global__ void k_dinv(float* __restrict__ deg) {
    int i = blockIdx.x * blockDim.x + threadIdx.x;
    if (i < NNODES) deg[i] = rsqrtf(deg[i]);   // deg >= 1 always (self loop)
}

// ---------------------------------------------------------------------------
// WMMA f32 GEMM: Out[M,NCOLS] = A[M,K] @ W[K,NCOLS] (+ bias)
// One wave per 16x16 tile. M is a multiple of 16 (50000 = 3125*16).
// Fragment layouts per CDNA5 ISA 7.12.2 (32-bit A 16x4, 32-bit C/D 16x16).
// ---------------------------------------------------------------------------
template <int K, int NCOLS, bool HASBIAS>
__global__ void k_gemm_wmma(const float* __restrict__ A,
                            const float* __restrict__ W,
                            const float* __restrict__ bias,
                            float* __restrict__ Out) {
    const int lane = threadIdx.x;        // 0..31, full wave active
    const int mt   = blockIdx.x;
    const int nt   = blockIdx.y;
    const int l15  = lane & 15;
    const int half = lane >> 4;          // 0: K pair {0,1}, 1: K pair {2,3}
    const int row  = mt * 16 + l15;      // A row for this lane
    const int col  = nt * 16 + l15;      // B/D column for this lane
    const bool colok = (NCOLS % 16 == 0) || (col < NCOLS);

    v8f acc = {};
    for (int k0 = 0; k0 < K; k0 += 4) {
        const int ka = k0 + half * 2;
        v2f a, b;
        a.x = A[row * K + ka];
        a.y = A[row * K + ka + 1];
        if (colok) {
            b.x = W[(ka    ) * NCOLS + col];
            b.y = W[(ka + 1) * NCOLS + col];
        } else {
            b.x = 0.0f; b.y = 0.0f;
        }
        acc = __builtin_amdgcn_wmma_f32_16x16x4_f32(
            /*neg_a=*/false, a, /*neg_b=*/false, b,
            /*c_mod=*/(short)0, acc, /*reuse_a=*/false, /*reuse_b=*/false);
    }

    const float bv = (HASBIAS && colok) ? bias[col] : 0.0f;
    const int orow0 = mt * 16 + half * 8;       // D: VGPR i holds row half*8 + i
#pragma unroll
    for (int i = 0; i < 8; ++i) {
        if (colok) Out[(orow0 + i) * NCOLS + col] = acc[i] + bv;
    }
}

// ---------------------------------------------------------------------------
// GCN aggregation
// ---------------------------------------------------------------------------
__global__ void k_selfloop(const float* __restrict__ xw,
                           const float* __restrict__ dinv,
                           float* __restrict__ h0) {
    int i = blockIdx.x * blockDim.x + threadIdx.x;
    if (i < NNODES * FILT) {
        float d = dinv[i / FILT];
        h0[i] = d * d * xw[i];
    }
}

// wave per edge, lane = feature (FILT == 32 == wave width)
__global__ void k_gcn_scatter(const int* __restrict__ row,
                              const int* __restrict__ col,
                              const float* __restrict__ dinv,
                              const float* __restrict__ xw,
                              float* __restrict__ h0) {
    int e = blockIdx.x * (blockDim.x >> 5) + (threadIdx.x >> 5);
    int f = threadIdx.x & 31;
    if (e >= NEDGES) return;
    int r = row[e], c = col[e];
    float nrm = dinv[r] * dinv[c];
    atomicAdd(&h0[c * FILT + f], nrm * xw[r * FILT + f]);
}

__global__ void k_relu_bias(float* __restrict__ h0, const float* __restrict__ b) {
    int i = blockIdx.x * blockDim.x + threadIdx.x;
    if (i < NNODES * FILT) h0[i] = fmaxf(h0[i] + b[i & 31], 0.0f);
}

// ---------------------------------------------------------------------------
// Attention
// ---------------------------------------------------------------------------
__device__ __forceinline__ void atomicMaxF32(float* addr, float v) {
    if (v >= 0.0f) atomicMax((int*)addr, __float_as_int(v));
    else           atomicMin((unsigned int*)addr, __float_as_uint(v));
}

// thread per (edge, head): score = q[col]·k[row] / sqrt(FILT)
__global__ void k_score(const int* __restrict__ row, const int* __restrict__ col,
                        const float* __restrict__ q, const float* __restrict__ k,
                        float* __restrict__ escore, float* __restrict__ smax) {
    int t = blockIdx.x * blockDim.x + threadIdx.x;
    if (t >= NEDGES * NHEADS) return;
    int e = t >> 2, h = t & 3;
    int r = row[e], c = col[e];
    const float* qp = q + c * HDIM + h * FILT;
    const float* kp = k + r * HDIM + h * FILT;
    float s = 0.0f;
#pragma unroll 8
    for (int j = 0; j < FILT; ++j) s += qp[j] * kp[j];
    s *= 0.17677669529663687f;   // 1/sqrt(32)
    escore[t] = s;
    atomicMaxF32(&smax[c * NHEADS + h], s);
}

__global__ void k_smax_fix(float* __restrict__ smax) {
    int i = blockIdx.x * blockDim.x + threadIdx.x;
    if (i < NNODES * NHEADS) {
        if (smax[i] == -__builtin_inff()) smax[i] = 0.0f;
    }
}

// ea = exp(score - smax[col]); denom[col] += ea   (ea overwrites escore)
__global__ void k_ea(const int* __restrict__ col, const float* __restrict__ smax,
                     float* __restrict__ ea, float* __restrict__ denom) {
    int t = blockIdx.x * blockDim.x + threadIdx.x;
    if (t >= NEDGES * NHEADS) return;
    int e = t >> 2, h = t & 3;
    int c = col[e];
    float v = expf(ea[t] - smax[c * NHEADS + h]);
    ea[t] = v;
    atomicAdd(&denom[c * NHEADS + h], v);
}

// wave per edge: hout[col] += alpha * v[row] across all 4 heads
__global__ void k_msg(const int* __restrict__ row, const int* __restrict__ col,
                      const float* __restrict__ ea, const float* __restrict__ denom,
                      const float* __restrict__ vmat, float* __restrict__ hout) {
    int e = blockIdx.x * (blockDim.x >> 5) + (threadIdx.x >> 5);
    int lane = threadIdx.x & 31;
    if (e >= NEDGES) return;
    int r = row[e], c = col[e];
#pragma unroll
    for (int h = 0; h < NHEADS; ++h) {
        float alpha = ea[e * NHEADS + h] / (denom[c * NHEADS + h] + 1e-16f);
        atomicAdd(&hout[c * HDIM + h * FILT + lane],
                  alpha * vmat[r * HDIM + h * FILT + lane]);
    }
}

// ---------------------------------------------------------------------------
extern "C" void kernel_launch(void* const* d_in, const int* in_sizes, int n_in,
                              void* d_out, int out_size, void* d_ws, size_t ws_size,
                              hipStream_t stream) {
    const float* x      = (const float*)d_in[0];
    const int*   ei     = (const int*)d_in[1];   // (2, E)
    const float* gcn_w  = (const float*)d_in[3];
    const float* gcn_b  = (const float*)d_in[4];
    const float* q_w    = (const float*)d_in[5];
    const float* q_b    = (const float*)d_in[6];
    const float* k_w    = (const float*)d_in[7];
    const float* k_b    = (const float*)d_in[8];
    const float* v_w    = (const float*)d_in[9];
    const float* v_b    = (const float*)d_in[10];
    const float* skip_w = (const float*)d_in[11];
    const float* skip_b = (const float*)d_in[12];
    const float* lin_w  = (const float*)d_in[13];
    const float* lin_b  = (const float*)d_in[14];

    const int* erow = ei;
    const int* ecol = ei + NEDGES;

    // outputs: (out [N,12], h [N,128]) concatenated
    float* out  = (float*)d_out;
    float* hout = out + (size_t)NNODES * OUTLEN;

    // workspace carve-up (floats)
    float* ws    = (float*)d_ws;
    float* dinv  = ws;                               // N     (deg -> dinv in place)
    float* xw    = dinv  + NNODES;                   // N*32
    float* h0    = xw    + (size_t)NNODES * FILT;    // N*32  (agg -> h0 in place)
    float* qb    = h0    + (size_t)NNODES * FILT;    // N*128
    float* kb    = qb    + (size_t)NNODES * HDIM;    // N*128
    float* vb    = kb    + (size_t)NNODES * HDIM;    // N*128
    float* smax  = vb    + (size_t)NNODES * HDIM;    // N*4
    float* denom = smax  + (size_t)NNODES * NHEADS;  // N*4
    float* ea    = denom + (size_t)NNODES * NHEADS;  // E*4

    const int MT = NNODES / 16;   // 3125, exact

    k_init<<<(NNODES * NHEADS + 255) / 256, 256, 0, stream>>>(dinv, smax, denom);
    k_degree<<<(NEDGES + 255) / 256, 256, 0, stream>>>(ecol, dinv);
    k_dinv<<<(NNODES + 255) / 256, 256, 0, stream>>>(dinv);

    // xw = x @ gcn_w
    k_gemm_wmma<NFIN, FILT, false><<<dim3(MT, FILT / 16), 32, 0, stream>>>(x, gcn_w, nullptr, xw);

    k_selfloop<<<(NNODES * FILT + 255) / 256, 256, 0, stream>>>(xw, dinv, h0);
    k_gcn_scatter<<<(NEDGES + 7) / 8, 256, 0, stream>>>(erow, ecol, dinv, xw, h0);
    k_relu_bias<<<(NNODES * FILT + 255) / 256, 256, 0, stream>>>(h0, gcn_b);

    // projections (skip writes the h output region directly; msg accumulates on top)
    k_gemm_wmma<FILT, HDIM, true><<<dim3(MT, HDIM / 16), 32, 0, stream>>>(h0, q_w, q_b, qb);
    k_gemm_wmma<FILT, HDIM, true><<<dim3(MT, HDIM / 16), 32, 0, stream>>>(h0, k_w, k_b, kb);
    k_gemm_wmma<FILT, HDIM, true><<<dim3(MT, HDIM / 16), 32, 0, stream>>>(h0, v_w, v_b, vb);
    k_gemm_wmma<FILT, HDIM, true><<<dim3(MT, HDIM / 16), 32, 0, stream>>>(h0, skip_w, skip_b, hout);

    // attention softmax over incoming edges
    k_score<<<(NEDGES * NHEADS + 255) / 256, 256, 0, stream>>>(erow, ecol, qb, kb, ea, smax);
    k_smax_fix<<<(NNODES * NHEADS + 255) / 256, 256, 0, stream>>>(smax);
    k_ea<<<(NEDGES * NHEADS + 255) / 256, 256, 0, stream>>>(ecol, smax, ea, denom);
    k_msg<<<(NEDGES + 7) / 8, 256, 0, stream>>>(erow, ecol, ea, denom, vb, hout);

    // out = h @ lin_w + lin_b
    k_gemm_wmma<HDIM, OUTLEN, true><<<dim3(MT, 1), 32, 0, stream>>>(hout, lin_w, lin_b, out);
}